// SymmetricContraction_25082609009189
// MI455X (gfx1250) — compile-verified
//
#include <hip/hip_runtime.h>

// ---------------------------------------------------------------------------
// MACE symmetric contraction (corr=3) for MI455X / gfx1250.
//   B=16384 nodes, C=64 channels, I=9, S=10 species, 4 outputs per (b,c).
// Strategy:
//   1) bucket nodes by species (atomic, order-independent)
//   2) precompute per-(e,c) folded tensors:
//        GA : A3 stacked (324x9 -> padded 336x12), pre-swizzled into the
//             v_wmma_f32_16x16x4_f32 A-operand register layout
//        A2 : 336 (padded) per (e,c)  -> used as WMMA C-matrix init
//        A1 : 36 per (e,c)
//   3) main kernel: per (32-node tile, channel) block
//        stage 1:  Out1(336x32) = G(336x12) x X(12x32) via WMMA f32 16x16x4,
//                  C initialized with A2 (folds the corr-2 add for free)
//        stage 2/3: per-node VALU fold to 4 outputs
// ---------------------------------------------------------------------------

typedef __attribute__((ext_vector_type(2))) float v2f;
typedef __attribute__((ext_vector_type(8))) float v8f;

#define NB      16384
#define NSP     10
#define NC      64
#define IDIM    9
#define K3      23
#define K2      5
#define K1      2
#define GROWS   336      // 324 (= 4 outputs * 81) padded to 21*16
#define GK      12       // 9 padded to 3*4
#define MTILES  21       // GROWS/16
#define TPB     256
#define TILE_B  32       // nodes per block (2 WMMA N-tiles)
#define MAXT    528      // worst-case species tiles of 32: 512 + 10 partials
#define RS      340      // O1 row stride (floats): 336 + 4 pad (bank spread)
#define GASZ    (MTILES * 3 * 32 * 2)   // 4032 floats per (e,c)

// ---------------------------------------------------------------- bucketing
__global__ void mace_bucket_init(int* counts, int* cursor, int* nTiles) {
    int t = threadIdx.x;
    if (t < 16) { counts[t] = 0; cursor[t] = 0; }
    if (t == 0) nTiles[0] = 0;
}

__global__ void mace_species(const float* __restrict__ y, int* __restrict__ sp,
                             int* __restrict__ counts) {
    int b = blockIdx.x * blockDim.x + threadIdx.x;
    if (b >= NB) return;
    int e = 0; float best = y[b * NSP];
    #pragma unroll
    for (int j = 1; j < NSP; ++j) {
        float v = y[b * NSP + j];
        if (v > best) { best = v; e = j; }
    }
    sp[b] = e;
    atomicAdd(&counts[e], 1);
}

__global__ void mace_scan_tiles(const int* __restrict__ counts, int* __restrict__ basev,
                                int* __restrict__ tile_sp, int* __restrict__ tile_base,
                                int* __restrict__ tile_cnt, int* __restrict__ nTiles) {
    if (threadIdx.x != 0 || blockIdx.x != 0) return;
    int acc = 0, T = 0;
    for (int e = 0; e < NSP; ++e) {
        basev[e] = acc;
        int cnt = counts[e];
        for (int s = 0; s < cnt; s += TILE_B) {
            tile_sp[T]   = e;
            tile_base[T] = acc + s;
            tile_cnt[T]  = (cnt - s < TILE_B) ? (cnt - s) : TILE_B;
            ++T;
        }
        acc += cnt;
    }
    nTiles[0] = T;
}

__global__ void mace_scatter(const int* __restrict__ sp, const int* __restrict__ basev,
                             int* __restrict__ cursor, int* __restrict__ node_index) {
    int b = blockIdx.x * blockDim.x + threadIdx.x;
    if (b >= NB) return;
    int e = sp[b];
    int p = atomicAdd(&cursor[e], 1);
    node_index[basev[e] + p] = b;
}

// ------------------------------------------------- per-(e,c) tensor folding
// Logical G[row, i], row = o*81 + i1*9 + i2 (o: 0=scalar, 1..3=vector comps).
// GA stored pre-swizzled for the WMMA A operand:
//   GA[((mt*3 + kc)*32 + lane)*2 + j] = G[mt*16 + (lane&15)][kc*4 + (lane>>4)*2 + j]
__global__ void mace_prep_A(const float* __restrict__ U3s, const float* __restrict__ U2s,
                            const float* __restrict__ U1s, const float* __restrict__ W3s,
                            const float* __restrict__ W2s, const float* __restrict__ W1s,
                            const float* __restrict__ U3v, const float* __restrict__ U2v,
                            const float* __restrict__ U1v, const float* __restrict__ W3v,
                            const float* __restrict__ W2v, const float* __restrict__ W1v,
                            float* __restrict__ GA, float* __restrict__ A2W,
                            float* __restrict__ A1W) {
    int ec = blockIdx.x;             // 0..639
    int e = ec / NC, c = ec % NC;
    int t = threadIdx.x;

    float* Gout = GA + (size_t)ec * GASZ;
    for (int idx = t; idx < GASZ; idx += TPB) {
        int j    = idx & 1;
        int lane = (idx >> 1) & 31;
        int kcmt = idx >> 6;
        int kc   = kcmt % 3;
        int mt   = kcmt / 3;
        int row  = mt * 16 + (lane & 15);
        int i    = kc * 4 + ((lane >> 4) << 1) + j;
        float v = 0.f;
        if (row < 324 && i < IDIM) {
            int o = row / 81, rr = row % 81;
            int i1 = rr / IDIM, i2 = rr % IDIM;
            if (o == 0) {
                const float* u = U3s + (size_t)((i1 * IDIM + i2) * IDIM + i) * K3;
                #pragma unroll
                for (int kk = 0; kk < K3; ++kk) v += u[kk] * W3s[(e * K3 + kk) * NC + c];
            } else {
                int a = o - 1;
                const float* u = U3v + (size_t)((((a * IDIM + i1) * IDIM + i2) * IDIM) + i) * K3;
                #pragma unroll
                for (int kk = 0; kk < K3; ++kk) v += u[kk] * W3v[(e * K3 + kk) * NC + c];
            }
        }
        Gout[idx] = v;
    }

    float* A2 = A2W + (size_t)ec * GROWS;        // padded to 336
    for (int idx = t; idx < GROWS; idx += TPB) {
        float v = 0.f;
        if (idx < 324) {
            int o = idx / 81, rr = idx % 81;
            if (o == 0) {
                const float* u = U2s + (size_t)rr * K2;
                #pragma unroll
                for (int kk = 0; kk < K2; ++kk) v += u[kk] * W2s[(e * K2 + kk) * NC + c];
            } else {
                const float* u = U2v + (size_t)((o - 1) * 81 + rr) * K2;
                #pragma unroll
                for (int kk = 0; kk < K2; ++kk) v += u[kk] * W2v[(e * K2 + kk) * NC + c];
            }
        }
        A2[idx] = v;
    }

    float* A1 = A1W + (size_t)ec * 36;
    if (t < 36) {
        int o = t / IDIM, i1 = t % IDIM;
        float v = 0.f;
        if (o == 0) {
            #pragma unroll
            for (int kk = 0; kk < K1; ++kk) v += U1s[i1 * K1 + kk] * W1s[(e * K1 + kk) * NC + c];
        } else {
            #pragma unroll
            for (int kk = 0; kk < K1; ++kk)
                v += U1v[(size_t)((o - 1) * IDIM + i1) * K1 + kk] * W1v[(e * K1 + kk) * NC + c];
        }
        A1[t] = v;
    }
}

// ------------------------------------------------------------- main kernel
__global__ __launch_bounds__(TPB)
void mace_main(const float* __restrict__ x, const float* __restrict__ GA,
               const float* __restrict__ A2W, const float* __restrict__ A1W,
               const int* __restrict__ node_index, const int* __restrict__ tile_sp,
               const int* __restrict__ tile_base, const int* __restrict__ tile_cnt,
               const int* __restrict__ nTiles, float* __restrict__ out) {
    __shared__ float Xs[GK * TILE_B];     // 1536 B : node features, padded K
    __shared__ float O1[TILE_B * RS];     // 43520 B: stage-1 results, col-major

    int tId = blockIdx.x;
    if (tId >= nTiles[0]) return;         // uniform across block
    int c    = blockIdx.y;
    int e    = tile_sp[tId];
    int base = tile_base[tId];
    int cnt  = tile_cnt[tId];
    int t    = threadIdx.x;

    const v2f*   GAp = (const v2f*)(GA + (size_t)(e * NC + c) * GASZ);
    const float* A2g = A2W + (size_t)(e * NC + c) * GROWS;
    const float* A1g = A1W + (size_t)(e * NC + c) * 36;

    for (int idx = t; idx < GK * TILE_B; idx += TPB) {   // X[k][n], zero-padded
        int k = idx / TILE_B, n = idx % TILE_B;
        float v = 0.f;
        if (k < IDIM && n < cnt) {
            int b = node_index[base + n];
            v = x[((size_t)b * NC + c) * IDIM + k];
        }
        Xs[idx] = v;
    }
    __syncthreads();

    // ---- stage 1: WMMA f32 16x16x4; C initialized with A2 (corr-2 add) ----
    int wave = t >> 5;                    // wave32
    int lane = t & 31;
    if (wave < 7) {                       // wave-uniform: EXEC stays all-ones
        int half = lane >> 4;             // K half: {0,1} vs {2,3}
        int l16  = lane & 15;
        v8f acc[3][2];
        #pragma unroll
        for (int mtl = 0; mtl < 3; ++mtl) {
            int row0 = (wave * 3 + mtl) * 16 + half * 8;   // 32B-aligned
            float4 q0 = *(const float4*)(A2g + row0);
            float4 q1 = *(const float4*)(A2g + row0 + 4);
            v8f ci;
            ci[0] = q0.x; ci[1] = q0.y; ci[2] = q0.z; ci[3] = q0.w;
            ci[4] = q1.x; ci[5] = q1.y; ci[6] = q1.z; ci[7] = q1.w;
            acc[mtl][0] = ci;
            acc[mtl][1] = ci;
        }
        #pragma unroll
        for (int kc = 0; kc < 3; ++kc) {
            int k0 = kc * 4 + half * 2;
            v2f b0, b1;
            b0.x = Xs[k0 * TILE_B + l16];       b0.y = Xs[(k0 + 1) * TILE_B + l16];
            b1.x = Xs[k0 * TILE_B + 16 + l16];  b1.y = Xs[(k0 + 1) * TILE_B + 16 + l16];
            #pragma unroll
            for (int mtl = 0; mtl < 3; ++mtl) {
                v2f a = GAp[((wave * 3 + mtl) * 3 + kc) * 32 + lane];  // coalesced b64
                acc[mtl][0] = __builtin_amdgcn_wmma_f32_16x16x4_f32(
                    false, a, false, b0, (short)0, acc[mtl][0], false, false);
                acc[mtl][1] = __builtin_amdgcn_wmma_f32_16x16x4_f32(
                    false, a, false, b1, (short)0, acc[mtl][1], false, false);
            }
        }
        // D writeback: lane holds rows (mt*16 + half*8 + 0..7) of col (nt*16+l16)
        #pragma unroll
        for (int mtl = 0; mtl < 3; ++mtl) {
            int row0 = (wave * 3 + mtl) * 16 + half * 8;
            #pragma unroll
            for (int nt = 0; nt < 2; ++nt) {
                int o1 = (nt * 16 + l16) * RS + row0;      // 16B-aligned
                float4 s0, s1;
                s0.x = acc[mtl][nt][0]; s0.y = acc[mtl][nt][1];
                s0.z = acc[mtl][nt][2]; s0.w = acc[mtl][nt][3];
                s1.x = acc[mtl][nt][4]; s1.y = acc[mtl][nt][5];
                s1.z = acc[mtl][nt][6]; s1.w = acc[mtl][nt][7];
                *(float4*)&O1[o1]     = s0;                // ds_store_b128
                *(float4*)&O1[o1 + 4] = s1;                // ds_store_b128
            }
        }
    }
    __syncthreads();

    // ---- stages 2+3: per-(node, output) VALU fold -------------------------
    if (t < TILE_B * 4) {
        int n = t & (TILE_B - 1), o = t >> 5;
        if (n < cnt) {
            float xr[IDIM];
            #pragma unroll
            for (int i = 0; i < IDIM; ++i) xr[i] = Xs[i * TILE_B + n];
            const float* row = &O1[n * RS + o * 81];       // already has A2 added
            float acc = 0.f;
            #pragma unroll
            for (int i1 = 0; i1 < IDIM; ++i1) {
                float s2 = 0.f;
                #pragma unroll
                for (int i2 = 0; i2 < IDIM; ++i2)
                    s2 += row[i1 * IDIM + i2] * xr[i2];
                acc += (s2 + A1g[o * IDIM + i1]) * xr[i1];
            }
            int b = node_index[base + n];
            if (o == 0) out[(size_t)b * 256 + c] = acc;                    // scalar block
            else        out[(size_t)b * 256 + 64 + c * 3 + (o - 1)] = acc; // vector block
        }
    }
}

// ---------------------------------------------------------------- launcher
extern "C" void kernel_launch(void* const* d_in, const int* in_sizes, int n_in,
                              void* d_out, int out_size, void* d_ws, size_t ws_size,
                              hipStream_t stream) {
    const float* x   = (const float*)d_in[0];
    const float* y   = (const float*)d_in[1];
    const float* U3s = (const float*)d_in[2];
    const float* U2s = (const float*)d_in[3];
    const float* U1s = (const float*)d_in[4];
    const float* W3s = (const float*)d_in[5];
    const float* W2s = (const float*)d_in[6];
    const float* W1s = (const float*)d_in[7];
    const float* U3v = (const float*)d_in[8];
    const float* U2v = (const float*)d_in[9];
    const float* U1v = (const float*)d_in[10];
    const float* W3v = (const float*)d_in[11];
    const float* W2v = (const float*)d_in[12];
    const float* W1v = (const float*)d_in[13];
    float* out = (float*)d_out;

    char* p = (char*)d_ws;
    float* GA  = (float*)p; p += (size_t)NSP * NC * GASZ * 4;    // 10.3 MB
    float* A2W = (float*)p; p += (size_t)NSP * NC * GROWS * 4;
    float* A1W = (float*)p; p += (size_t)NSP * NC * 36 * 4;
    int* counts     = (int*)p; p += 16 * 4;
    int* cursor     = (int*)p; p += 16 * 4;
    int* basev      = (int*)p; p += 16 * 4;
    int* nTiles     = (int*)p; p += 16 * 4;
    int* sp         = (int*)p; p += (size_t)NB * 4;
    int* node_index = (int*)p; p += (size_t)NB * 4;
    int* tile_sp    = (int*)p; p += (size_t)(MAXT + 8) * 4;
    int* tile_base  = (int*)p; p += (size_t)(MAXT + 8) * 4;
    int* tile_cnt   = (int*)p; p += (size_t)(MAXT + 8) * 4;

    mace_bucket_init<<<1, 32, 0, stream>>>(counts, cursor, nTiles);
    mace_species<<<NB / TPB, TPB, 0, stream>>>(y, sp, counts);
    mace_scan_tiles<<<1, 1, 0, stream>>>(counts, basev, tile_sp, tile_base, tile_cnt, nTiles);
    mace_scatter<<<NB / TPB, TPB, 0, stream>>>(sp, basev, cursor, node_index);
    mace_prep_A<<<NSP * NC, TPB, 0, stream>>>(U3s, U2s, U1s, W3s, W2s, W1s,
                                              U3v, U2v, U1v, W3v, W2v, W1v,
                                              GA, A2W, A1W);
    dim3 grid(MAXT, NC);
    mace_main<<<grid, TPB, 0, stream>>>(x, GA, A2W, A1W, node_index,
                                        tile_sp, tile_base, tile_cnt, nTiles, out);
}